// RGate_56573309222986
// MI455X (gfx1250) — compile-verified
//
#include <hip/hip_runtime.h>

typedef float v2f __attribute__((ext_vector_type(2)));
typedef float v8f __attribute__((ext_vector_type(8)));

#define NSTATE (1 << 22)
#define TILE1  16384     // pass-1 tile: bits 0..13
#define T1     512       // pass-1 threads (16 waves)
#define CH     32        // pass-2 lo-chunk width
#define PITCH  33        // pass-2 LDS row pitch (odd -> conflict-free A loads)
#define T2     512

// D = A(16x4) * B(4x16) + C(16x16), f32 WMMA
__device__ __forceinline__ v8f wmma4(v2f a, v2f b, v8f c) {
  return __builtin_amdgcn_wmma_f32_16x16x4_f32(false, a, false, b, (short)0, c,
                                               false, false);
}

// ---------------------------------------------------------------------------
// Setup: build 16x16 complex group matrices G = kron of four 2x2 gates.
// ws layout (floats):
//   m*512 + [0..255]   : Gr[m] (row-major 16x16)
//   m*512 + [256..511] : Gi[m]
//   2048 + 2p, 2048+2p+1 : cos/sin(angle[21-p]/2) for flat bit p
// Groups: m=0 bits0-3, m=1 bits4-7, m=2 bits14-17, m=3 bits18-21.
// Flat bit p uses angle[21-p] (reshape axis i == flat bit 21-i).
// ---------------------------------------------------------------------------
__global__ void setup_kernel(const float* __restrict__ angle,
                             float* __restrict__ ws) {
  const int tid = threadIdx.x;           // 256 threads
  const int gp = tid >> 4, g = tid & 15; // row g', col g
  const int diff = gp ^ g;
  const int bases[4] = {0, 4, 14, 18};
#pragma unroll
  for (int m = 0; m < 4; ++m) {
    float mag = 1.0f;
#pragma unroll
    for (int j = 0; j < 4; ++j) {
      float th = 0.5f * angle[21 - (bases[m] + j)];
      mag *= ((diff >> j) & 1) ? sinf(th) : cosf(th);
    }
    int pc = __popc(diff) & 3;  // phase (-i)^pc
    float re = (pc == 0) ? mag : ((pc == 2) ? -mag : 0.0f);
    float im = (pc == 1) ? -mag : ((pc == 3) ? mag : 0.0f);
    ws[m * 512 + tid]       = re;
    ws[m * 512 + 256 + tid] = im;
  }
  if (tid < 22) {
    float th = 0.5f * angle[21 - tid];
    ws[2048 + 2 * tid]     = cosf(th);
    ws[2048 + 2 * tid + 1] = sinf(th);
  }
}

// Per-wave constant operand registers.
struct Mats {
  v2f Br[4], Bi[4], nBi[4];  // B-layout of G_B^T (== A-indexing of G_B)
  v2f Ar[4], Ai[4], nAi[4];  // A-layout of G_A, K-slices reordered (2k,2k+1,2k+8,2k+9)
};

__device__ __forceinline__ void load_consts(const float* gBr, const float* gBi,
                                            const float* gAr, const float* gAi,
                                            int lane, Mats& M) {
  const int row = lane & 15, half = lane >> 4;
#pragma unroll
  for (int k = 0; k < 4; ++k) {
#pragma unroll
    for (int j = 0; j < 2; ++j) {
      const int cB = 4 * k + j + 2 * half;   // transpose cancels: index G_B directly
      M.Br[k][j]  = gBr[row * 16 + cB];
      M.Bi[k][j]  = gBi[row * 16 + cB];
      M.nBi[k][j] = -M.Bi[k][j];
      const int cA = 2 * k + j + 8 * half;   // reordered K so Z-acc pairs feed B directly
      M.Ar[k][j]  = gAr[row * 16 + cA];
      M.Ai[k][j]  = gAi[row * 16 + cA];
      M.nAi[k][j] = -M.Ai[k][j];
    }
  }
}

// Y = G_A * X * G_B^T (complex), X given as A-layout slices. Y left in C/D layout.
__device__ __forceinline__ void block_wmma(const Mats& M, const v2f Xr[4],
                                           const v2f Xi[4], v8f& yr, v8f& yi) {
  v8f zr = {}, zi = {};
#pragma unroll
  for (int k = 0; k < 4; ++k) zr = wmma4(Xr[k], M.Br[k], zr);
#pragma unroll
  for (int k = 0; k < 4; ++k) zr = wmma4(Xi[k], M.nBi[k], zr);
#pragma unroll
  for (int k = 0; k < 4; ++k) zi = wmma4(Xr[k], M.Bi[k], zi);
#pragma unroll
  for (int k = 0; k < 4; ++k) zi = wmma4(Xi[k], M.Br[k], zi);
  // Step 2: Z acc VGPR pairs (2k,2k+1) are directly valid B slices.
#pragma unroll
  for (int k = 0; k < 4; ++k) {
    v2f b; b[0] = zr[2 * k]; b[1] = zr[2 * k + 1];
    yr = wmma4(M.Ar[k], b, yr);
  }
#pragma unroll
  for (int k = 0; k < 4; ++k) {
    v2f b; b[0] = zi[2 * k]; b[1] = zi[2 * k + 1];
    yr = wmma4(M.nAi[k], b, yr);
  }
#pragma unroll
  for (int k = 0; k < 4; ++k) {
    v2f b; b[0] = zi[2 * k]; b[1] = zi[2 * k + 1];
    yi = wmma4(M.Ar[k], b, yi);
  }
#pragma unroll
  for (int k = 0; k < 4; ++k) {
    v2f b; b[0] = zr[2 * k]; b[1] = zr[2 * k + 1];
    yi = wmma4(M.Ai[k], b, yi);
  }
}

// ---------------------------------------------------------------------------
// Pass 1: contiguous tile of 2^14. Butterflies on bits 8..13, WMMA on bits 0..7.
// ---------------------------------------------------------------------------
__global__ __launch_bounds__(T1) void pass1_kernel(
    const float* __restrict__ xr, const float* __restrict__ xi,
    const float* __restrict__ ws, float* __restrict__ outr,
    float* __restrict__ outi) {
  __shared__ float Lr[TILE1];
  __shared__ float Li[TILE1];
  const int tid  = threadIdx.x;
  const int base = blockIdx.x * TILE1;

  for (int i = tid; i < TILE1; i += T1) {
    Lr[i] = xr[base + i];
    Li[i] = xi[base + i];
  }
  __syncthreads();

  // 6 butterfly stages: flat bits 8..13
  for (int b = 8; b <= 13; ++b) {
    const float c = ws[2048 + 2 * b], s = ws[2048 + 2 * b + 1];
    const int mask = (1 << b) - 1;
    for (int p = tid; p < TILE1 / 2; p += T1) {
      const int a = ((p & ~mask) << 1) | (p & mask);
      const int q = a | (1 << b);
      float ar = Lr[a], ai = Li[a], br = Lr[q], bi = Li[q];
      Lr[a] = c * ar + s * bi;  Li[a] = c * ai - s * br;
      Lr[q] = c * br + s * ai;  Li[q] = c * bi - s * ar;
    }
    __syncthreads();
  }

  // WMMA: each 256-block, X[g1=bits4-7][g0=bits0-3], Y = G1 * X * G0^T
  const int lane = tid & 31, wave = tid >> 5;
  const int row = lane & 15, half = lane >> 4;
  Mats M;
  load_consts(ws + 0, ws + 256, ws + 512, ws + 768, lane, M);

  for (int blk = wave; blk < TILE1 / 256; blk += T1 / 32) {
    const int bb = blk * 256;
    v2f Xr[4], Xi[4];
#pragma unroll
    for (int k = 0; k < 4; ++k) {
      const int addr = bb + row * 16 + 4 * k + 2 * half;  // 8B aligned
      Xr[k] = *(const v2f*)&Lr[addr];
      Xi[k] = *(const v2f*)&Li[addr];
    }
    v8f yr = {}, yi = {};
    block_wmma(M, Xr, Xi, yr, yi);
#pragma unroll
    for (int v = 0; v < 8; ++v) {  // C/D layout -> n = g0' + 16*g1'
      const int n = row + 16 * v + 128 * half;
      outr[base + bb + n] = yr[v];
      outi[base + bb + n] = yi[v];
    }
  }
}

// ---------------------------------------------------------------------------
// Pass 2: bits 14..21, pure WMMA, in place on d_out.
// LDS tile stored permuted: hrow = (hi&15)*16 + (hi>>4)  ([g0][g1] order),
// addr = hrow*PITCH + lo  -> conflict-free A-operand loads.
// ---------------------------------------------------------------------------
__global__ __launch_bounds__(T2) void pass2_kernel(const float* __restrict__ ws,
                                                   float* __restrict__ outr,
                                                   float* __restrict__ outi) {
  __shared__ float Lr[256 * PITCH];
  __shared__ float Li[256 * PITCH];
  const int tid    = threadIdx.x;
  const int lobase = blockIdx.x * CH;

  for (int t = tid; t < 256 * CH; t += T2) {
    const int hi = t >> 5;           // CH == 32
    const int lo = t & (CH - 1);
    const int hrow = ((hi & 15) << 4) | (hi >> 4);
    const int gidx = (lobase + lo) + (hi << 14);
    Lr[hrow * PITCH + lo] = outr[gidx];
    Li[hrow * PITCH + lo] = outi[gidx];
  }
  __syncthreads();

  const int lane = tid & 31, wave = tid >> 5;
  const int row = lane & 15, half = lane >> 4;
  Mats M;
  load_consts(ws + 1024, ws + 1280, ws + 1536, ws + 1792, lane, M);

  // Each wave owns lo columns {wave, wave+16}: no inter-wave hazard.
  for (int lo = wave; lo < CH; lo += T2 / 32) {
    v2f Xr[4], Xi[4];
#pragma unroll
    for (int k = 0; k < 4; ++k) {
      const int g0a = 4 * k + 2 * half;  // X[M=g1=row][K=g0]
      const int a0 = (g0a * 16 + row) * PITCH + lo;
      const int a1 = ((g0a + 1) * 16 + row) * PITCH + lo;
      Xr[k][0] = Lr[a0]; Xr[k][1] = Lr[a1];
      Xi[k][0] = Li[a0]; Xi[k][1] = Li[a1];
    }
    v8f yr = {}, yi = {};
    block_wmma(M, Xr, Xi, yr, yi);
#pragma unroll
    for (int v = 0; v < 8; ++v) {  // Y[g1'=v+8h][g0'=row] -> hrow' = g0'*16+g1'
      const int a = (row * 16 + v + 8 * half) * PITCH + lo;
      Lr[a] = yr[v];
      Li[a] = yi[v];
    }
  }
  __syncthreads();

  for (int t = tid; t < 256 * CH; t += T2) {
    const int hi = t >> 5;
    const int lo = t & (CH - 1);
    const int hrow = ((hi & 15) << 4) | (hi >> 4);
    const int gidx = (lobase + lo) + (hi << 14);
    outr[gidx] = Lr[hrow * PITCH + lo];
    outi[gidx] = Li[hrow * PITCH + lo];
  }
}

extern "C" void kernel_launch(void* const* d_in, const int* in_sizes, int n_in,
                              void* d_out, int out_size, void* d_ws,
                              size_t ws_size, hipStream_t stream) {
  const float* xr    = (const float*)d_in[0];
  const float* xi    = (const float*)d_in[1];
  const float* angle = (const float*)d_in[2];
  float* outr = (float*)d_out;
  float* outi = (float*)d_out + NSTATE;
  float* ws   = (float*)d_ws;

  setup_kernel<<<1, 256, 0, stream>>>(angle, ws);
  pass1_kernel<<<NSTATE / TILE1, T1, 0, stream>>>(xr, xi, ws, outr, outi);
  pass2_kernel<<<(1 << 14) / CH, T2, 0, stream>>>(ws, outr, outi);
}